// OrkyDocumentRetriever_72164040507671
// MI455X (gfx1250) — compile-verified
//
#include <hip/hip_runtime.h>
#include <math.h>

#define B_Q    32
#define N_DOCS 16384
#define S_LEN  32
#define D_DIM  256
#define TOPK   5
#define EPS_F  1e-8f

typedef __attribute__((ext_vector_type(2))) float v2f;
typedef __attribute__((ext_vector_type(4))) float v4f;
typedef __attribute__((ext_vector_type(8))) float v8f;

// D = A(16x4 f32) * B(4x16 f32) + C(16x16 f32), wave32
__device__ __forceinline__ v8f wmma_f32_16x16x4(v2f a, v2f b, v8f c) {
    return __builtin_amdgcn_wmma_f32_16x16x4_f32(
        /*neg_a=*/false, a, /*neg_b=*/false, b,
        /*c_mod=*/(short)0, c, /*reuse_a=*/false, /*reuse_b=*/false);
}

// ---------------------------------------------------------------------------
// Stage 1: doc_mean[n][d] = mean_s docs[n][s][d]
// The dominant 512MB HBM stream: b128 non-temporal loads (docs seen once,
// don't pollute L2 which must keep doc_mean/dth/sims hot for later stages).
// grid = N_DOCS/4 blocks, block = 256 thr; each wave-pair (64 thr) owns a doc,
// each thread owns 4 consecutive d's -> 512B per wave per load instruction.
__global__ void mean_kernel(const float* __restrict__ docs,
                            float* __restrict__ doc_mean) {
    const int tid = threadIdx.x;
    const int n   = blockIdx.x * 4 + (tid >> 6);   // 4 docs per block
    const int d4  = tid & 63;                      // float4 index within row
    const v4f* __restrict__ p =
        (const v4f*)(docs + (size_t)n * (S_LEN * D_DIM)) + d4;
    v4f acc = {};
#pragma unroll
    for (int s = 0; s < S_LEN; ++s) {
        v4f v = __builtin_nontemporal_load(p + s * (D_DIM / 4));
        acc += v;
    }
    acc *= (1.0f / (float)S_LEN);
    *((v4f*)(doc_mean + (size_t)n * D_DIM) + d4) = acc;
}

// ---------------------------------------------------------------------------
// Stage 2a: q = query @ Wq^T + bq   [32 x 256], grid 16 (n-tiles) x 64 thr (2 waves)
__global__ void qproj_kernel(const float* __restrict__ query,
                             const float* __restrict__ Wq,
                             const float* __restrict__ bq,
                             float* __restrict__ q) {
    const int lane = threadIdx.x & 31;
    const int w    = threadIdx.x >> 5;          // m-tile 0..1
    const int m0   = w * 16;
    const int n0   = blockIdx.x * 16;
    const int row  = m0 + (lane & 15);
    const int ncol = n0 + (lane & 15);
    const int ksel = (lane >> 4) << 1;          // lanes 16-31 handle K+2,K+3
    v8f acc = {};
    for (int k = 0; k < D_DIM; k += 4) {
        const int kk = k + ksel;
        v2f a = *(const v2f*)(query + (size_t)row * D_DIM + kk);
        v2f b = *(const v2f*)(Wq + (size_t)ncol * D_DIM + kk); // B[k][n] = Wq[n][k]
        acc = wmma_f32_16x16x4(a, b, acc);
    }
    const float bias = bq[ncol];
    const int rbase = m0 + (lane >> 4) * 8;
#pragma unroll
    for (int r = 0; r < 8; ++r)
        q[(size_t)(rbase + r) * D_DIM + ncol] = acc[r] + bias;
}

// ---------------------------------------------------------------------------
// Stage 2b: dth = doc_mean @ Wd^T + bd  [16384 x 256]
// grid (128, 16), block 256 (8 waves stacked in M). Wd n-slice staged in LDS.
// doc_mean re-reads across blockIdx.y are served from L2 (16MB fits easily).
__global__ void dproj_kernel(const float* __restrict__ doc_mean,
                             const float* __restrict__ Wd,
                             const float* __restrict__ bd,
                             float* __restrict__ dth) {
    __shared__ float Bs[D_DIM * 16];            // Bs[k*16+nn] = Wd[(n0+nn)][k]
    const int tid = threadIdx.x;
    const int n0  = blockIdx.y * 16;
    {
        const int k = tid;                      // 0..255
#pragma unroll
        for (int nn = 0; nn < 16; ++nn)
            Bs[k * 16 + nn] = Wd[(size_t)(n0 + nn) * D_DIM + k];
    }
    __syncthreads();

    const int lane = tid & 31;
    const int w    = tid >> 5;
    const int m0   = blockIdx.x * 128 + w * 16;
    const int row  = m0 + (lane & 15);
    const int nn   = lane & 15;
    const int ksel = (lane >> 4) << 1;
    v8f acc = {};
    for (int k = 0; k < D_DIM; k += 4) {
        const int kk = k + ksel;
        v2f a = *(const v2f*)(doc_mean + (size_t)row * D_DIM + kk);
        v2f b;
        b.x = Bs[(kk + 0) * 16 + nn];
        b.y = Bs[(kk + 1) * 16 + nn];
        acc = wmma_f32_16x16x4(a, b, acc);
    }
    const int   ncol = n0 + nn;
    const float bias = bd[ncol];
    const int rbase = m0 + (lane >> 4) * 8;
#pragma unroll
    for (int r = 0; r < 8; ++r)
        dth[(size_t)(rbase + r) * D_DIM + ncol] = acc[r] + bias;
}

// ---------------------------------------------------------------------------
// Norms: one wave per row, shfl reduction
__global__ void qnorm_kernel(const float* __restrict__ q, float* __restrict__ qn) {
    const int lane = threadIdx.x & 31;
    const int w    = threadIdx.x >> 5;          // 8 waves, 4 rows each
    for (int i = 0; i < 4; ++i) {
        const int r = w * 4 + i;
        float s = 0.f;
#pragma unroll
        for (int j = 0; j < 8; ++j) {
            float x = q[(size_t)r * D_DIM + lane + j * 32];
            s += x * x;
        }
        for (int off = 16; off > 0; off >>= 1) s += __shfl_down(s, off, 32);
        if (lane == 0) qn[r] = sqrtf(s);
    }
}

__global__ void dnorm_kernel(const float* __restrict__ dth, float* __restrict__ dn) {
    const int lane = threadIdx.x & 31;
    const int w    = threadIdx.x >> 5;
    const int r    = blockIdx.x * 8 + w;
    float s = 0.f;
#pragma unroll
    for (int j = 0; j < 8; ++j) {
        float x = dth[(size_t)r * D_DIM + lane + j * 32];
        s += x * x;
    }
    for (int off = 16; off > 0; off >>= 1) s += __shfl_down(s, off, 32);
    if (lane == 0) dn[r] = sqrtf(s);
}

// ---------------------------------------------------------------------------
// Stage 3: sims[b][n] = (q @ dth^T) / max(qn*dn, eps)   [32 x 16384]
// 2048 tiles; grid 256 x 256 thr (8 waves, one tile each)
__global__ void sims_kernel(const float* __restrict__ q,
                            const float* __restrict__ dth,
                            const float* __restrict__ qn,
                            const float* __restrict__ dn,
                            float* __restrict__ sims) {
    const int tid  = threadIdx.x;
    const int lane = tid & 31;
    const int w    = tid >> 5;
    const int tile = blockIdx.x * 8 + w;
    const int m0   = (tile & 1) * 16;
    const int n0   = (tile >> 1) * 16;
    const int row  = m0 + (lane & 15);
    const int ncol = n0 + (lane & 15);
    const int ksel = (lane >> 4) << 1;
    v8f acc = {};
    for (int k = 0; k < D_DIM; k += 4) {
        const int kk = k + ksel;
        v2f a = *(const v2f*)(q + (size_t)row * D_DIM + kk);
        v2f b = *(const v2f*)(dth + (size_t)ncol * D_DIM + kk); // B[k][n]=dth[n][k]
        acc = wmma_f32_16x16x4(a, b, acc);
    }
    const float dnv = dn[ncol];
    const int rbase = m0 + (lane >> 4) * 8;
#pragma unroll
    for (int r = 0; r < 8; ++r) {
        const int rq = rbase + r;
        sims[(size_t)rq * N_DOCS + ncol] = acc[r] / fmaxf(qn[rq] * dnv, EPS_F);
    }
}

// ---------------------------------------------------------------------------
// Top-5 per query. Per-thread sorted top-5 over strided chunk, LDS tree merge.
// Ties resolved toward lower index (matches lax.top_k).
__global__ void topk_kernel(const float* __restrict__ sims,
                            float* __restrict__ out_scores,
                            int* __restrict__ out_idx) {
    const int b   = blockIdx.x;
    const int tid = threadIdx.x;
    __shared__ float sv[256 * TOPK];
    __shared__ int   si[256 * TOPK];

    float v[TOPK]; int ix[TOPK];
#pragma unroll
    for (int i = 0; i < TOPK; ++i) { v[i] = -INFINITY; ix[i] = 0x7fffffff; }

    const float* row = sims + (size_t)b * N_DOCS;
    for (int n = tid; n < N_DOCS; n += 256) {
        const float s = row[n];
        if (s > v[TOPK - 1]) {                 // strict > keeps earlier index on tie
            int p = TOPK - 1;
            while (p > 0 && s > v[p - 1]) { v[p] = v[p - 1]; ix[p] = ix[p - 1]; --p; }
            v[p] = s; ix[p] = n;
        }
    }
#pragma unroll
    for (int i = 0; i < TOPK; ++i) { sv[tid * TOPK + i] = v[i]; si[tid * TOPK + i] = ix[i]; }
    __syncthreads();

    for (int stride = 128; stride > 0; stride >>= 1) {
        if (tid < stride) {
            const int a0 = tid * TOPK, b0 = (tid + stride) * TOPK;
            float rv[TOPK]; int ri[TOPK];
            int ia = 0, ib = 0;
#pragma unroll
            for (int k = 0; k < TOPK; ++k) {
                const float av = sv[a0 + ia], bv = sv[b0 + ib];
                const int   ai = si[a0 + ia], bi = si[b0 + ib];
                const bool takeA = (av > bv) || (av == bv && ai <= bi);
                if (takeA) { rv[k] = av; ri[k] = ai; ++ia; }
                else       { rv[k] = bv; ri[k] = bi; ++ib; }
            }
#pragma unroll
            for (int k = 0; k < TOPK; ++k) { sv[a0 + k] = rv[k]; si[a0 + k] = ri[k]; }
        }
        __syncthreads();
    }
    if (tid < TOPK) {
        out_scores[b * TOPK + tid] = sv[tid];
        out_idx[b * TOPK + tid]    = si[tid];
    }
}

// ---------------------------------------------------------------------------
// Gather full retrieved docs: out[b][k] = docs[idx[b][k]]  (32KB per winner).
// NT loads: random one-shot re-read of docs, no reuse.
__global__ void gather_kernel(const float* __restrict__ docs,
                              const int* __restrict__ idx,
                              float* __restrict__ out) {
    const int t   = blockIdx.x;                // b*TOPK + k
    const int doc = idx[t];
    const v4f* __restrict__ src = (const v4f*)(docs + (size_t)doc * (S_LEN * D_DIM));
    v4f* __restrict__ dst       = (v4f*)(out + (size_t)t * (S_LEN * D_DIM));
    for (int i = threadIdx.x; i < (S_LEN * D_DIM) / 4; i += 256)
        dst[i] = __builtin_nontemporal_load(src + i);
}

// ---------------------------------------------------------------------------
extern "C" void kernel_launch(void* const* d_in, const int* in_sizes, int n_in,
                              void* d_out, int out_size, void* d_ws, size_t ws_size,
                              hipStream_t stream) {
    (void)in_sizes; (void)n_in; (void)out_size; (void)ws_size;
    const float* query = (const float*)d_in[0];   // [32, 256]
    const float* docs  = (const float*)d_in[1];   // [16384, 32, 256]
    const float* Wq    = (const float*)d_in[2];   // [256, 256]
    const float* bq    = (const float*)d_in[3];   // [256]
    const float* Wd    = (const float*)d_in[4];   // [256, 256]
    const float* bd    = (const float*)d_in[5];   // [256]
    float* out = (float*)d_out;                   // retrieved [32,5,32,256] ++ scores [32,5]

    // Workspace layout (floats): ~36 MB total
    float* ws       = (float*)d_ws;
    float* doc_mean = ws;                                   // 16384*256
    float* dth      = doc_mean + (size_t)N_DOCS * D_DIM;    // 16384*256
    float* qbuf     = dth + (size_t)N_DOCS * D_DIM;         // 32*256
    float* qn       = qbuf + B_Q * D_DIM;                   // 32
    float* dn       = qn + B_Q;                             // 16384
    float* sims     = dn + N_DOCS;                          // 32*16384
    int*   idx      = (int*)(sims + (size_t)B_Q * N_DOCS);  // 160

    mean_kernel <<<N_DOCS / 4, 256, 0, stream>>>(docs, doc_mean);
    qproj_kernel<<<16, 64, 0, stream>>>(query, Wq, bq, qbuf);
    dproj_kernel<<<dim3(128, 16), 256, 0, stream>>>(doc_mean, Wd, bd, dth);
    qnorm_kernel<<<1, 256, 0, stream>>>(qbuf, qn);
    dnorm_kernel<<<N_DOCS / 8, 256, 0, stream>>>(dth, dn);
    sims_kernel <<<256, 256, 0, stream>>>(qbuf, dth, qn, dn, sims);

    float* out_scores = out + (size_t)B_Q * TOPK * S_LEN * D_DIM;
    topk_kernel  <<<B_Q, 256, 0, stream>>>(sims, out_scores, idx);
    gather_kernel<<<B_Q * TOPK, 256, 0, stream>>>(docs, idx, out);
}